// PointNetModule_5506148074007
// MI455X (gfx1250) — compile-verified
//
#include <hip/hip_runtime.h>

// ---------------- problem constants ----------------
#define B_   4
#define N_   8192
#define S_   2048
#define KSL  32                    // ball-query slots (K)
#define CIN  16
#define P_   (B_*S_*KSL)           // 262144 grouped points
#define R2_  0.25f                 // radius^2
#define EPS_ 1e-5f

typedef __attribute__((ext_vector_type(2))) float v2f;
typedef __attribute__((ext_vector_type(8))) float v8f;

// ---------------- init: zero pad row of X0 and the stat buffers ----------------
__global__ __launch_bounds__(256) void k_init(float* __restrict__ X0,
                                              float* __restrict__ stats) {
    int i = blockIdx.x * 256 + threadIdx.x;
    if (i < P_) X0[19 * (size_t)P_ + i] = 0.0f;   // zero-padded K row for GEMM1
    if (i < 384) stats[i] = 0.0f;                 // st1(128) st2(128) st3(128)
}

// ---------------- ball query: one thread per query point ----------------
#define TILE_ 2048
__global__ __launch_bounds__(256) void k_ballquery(const float* __restrict__ pc,
                                                   const float* __restrict__ new_pc,
                                                   int* __restrict__ idx,
                                                   int* __restrict__ num) {
    __shared__ float lx[TILE_], ly[TILE_], lz[TILE_];
    const int tid = threadIdx.x;
    const int b  = blockIdx.x / (S_ / 256);
    const int s  = (blockIdx.x % (S_ / 256)) * 256 + tid;
    const int q  = b * S_ + s;

    const float qx = new_pc[(b * 3 + 0) * S_ + s];
    const float qy = new_pc[(b * 3 + 1) * S_ + s];
    const float qz = new_pc[(b * 3 + 2) * S_ + s];

    int total = 0;          // full in-radius count (reference num)
    int firstj = 0;         // first found index (fill value)

    for (int t = 0; t < N_; t += TILE_) {
        // stage a tile of pc into LDS
        for (int i = tid; i < TILE_; i += 256) {
            lx[i] = pc[(b * 3 + 0) * N_ + t + i];
            ly[i] = pc[(b * 3 + 1) * N_ + t + i];
            lz[i] = pc[(b * 3 + 2) * N_ + t + i];
        }
        if (t + TILE_ < N_)   // hint next tile into cache (global_prefetch_b8)
            __builtin_prefetch(&pc[(b * 3 + 0) * N_ + t + TILE_], 0, 0);
        __syncthreads();

        #pragma unroll 4
        for (int i = 0; i < TILE_; ++i) {
            const float dx = lx[i] - qx;
            const float dy = ly[i] - qy;
            const float dz = lz[i] - qz;
            const float d2 = dx * dx + dy * dy + dz * dz;
            if (d2 < R2_) {
                if (total == 0) firstj = t + i;
                if (total < KSL) idx[q * KSL + total] = t + i;
                ++total;
            }
        }
        __syncthreads();
    }
    num[q] = total;
    for (int k = (total < KSL ? total : KSL); k < KSL; ++k)
        idx[q * KSL + k] = firstj;   // firstj==0 when total==0 (matches reference)
}

// ---------------- gather + center + concat -> X0 [20][P] ----------------
__global__ __launch_bounds__(256) void k_gather(const float* __restrict__ pc,
                                                const float* __restrict__ feat,
                                                const float* __restrict__ new_pc,
                                                const int* __restrict__ idx,
                                                float* __restrict__ X0) {
    const int p = blockIdx.x * 256 + threadIdx.x;   // grouped-point index
    const int q = p / KSL;
    const int b = q / S_;
    const int s = q % S_;
    const int j = idx[p];

    X0[0 * (size_t)P_ + p] = pc[(b * 3 + 0) * N_ + j] - new_pc[(b * 3 + 0) * S_ + s];
    X0[1 * (size_t)P_ + p] = pc[(b * 3 + 1) * N_ + j] - new_pc[(b * 3 + 1) * S_ + s];
    X0[2 * (size_t)P_ + p] = pc[(b * 3 + 2) * N_ + j] - new_pc[(b * 3 + 2) * S_ + s];
    #pragma unroll
    for (int c = 0; c < CIN; ++c)
        X0[(3 + c) * (size_t)P_ + p] = feat[(b * CIN + c) * N_ + j];
}

// ---------------- GEMM via V_WMMA_F32_16X16X4_F32 ----------------
// Y[16*MT, P] = W[16*MT, KW] * X[KD(pad), P] + bias ; also accumulates per-channel
// sum / sumsq into stats for training-mode BatchNorm. TO_OUT scatters directly
// into the [B,64,S,K] output layout (layer 3).
// Weights staged zero-padded into LDS (branch-free ds_load_2addr_b64 A-frags).
// Stats go through a 16-slot-padded LDS array: each lane's ds_add_f32 targets a
// distinct address (slot = lane&15), avoiding same-address atomic serialization.
template <int MT, int KD, int KW, bool TO_OUT>
__global__ __launch_bounds__(256) void k_gemm(const float* __restrict__ X,
                                              const float* __restrict__ W,
                                              const float* __restrict__ bias,
                                              float* __restrict__ Y,
                                              float* __restrict__ stats) {
    const int M = 16 * MT;
    __shared__ float Wl[64 * KD];           // M*KD used, zero-padded K rows
    __shared__ float red[16 * 2 * 64];      // [slot][2*M] padded partial sums
    const int tid = threadIdx.x;

    for (int i = tid; i < M * KD; i += 256) {
        const int row = i / KD, kk = i % KD;
        Wl[i] = (kk < KW) ? W[row * KW + kk] : 0.0f;
    }
    for (int i = tid; i < 16 * 2 * M; i += 256) red[i] = 0.0f;
    __syncthreads();

    const int wave = tid >> 5;
    const int lane = tid & 31;
    const int n0   = (blockIdx.x * 8 + wave) * 16;
    const int col  = n0 + (lane & 15);
    const int h    = lane >> 4;
    const int slot = (lane & 15) * 2 * M;   // private stat column per lane-of-half

    v8f acc[MT] = {};

    for (int kb = 0; kb < KD; kb += 4) {
        const int ka = kb + 2 * h;
        v2f bf;
        bf.x = X[(size_t)(ka)     * P_ + col];
        bf.y = X[(size_t)(ka + 1) * P_ + col];
        #pragma unroll
        for (int mt = 0; mt < MT; ++mt) {
            const int m = mt * 16 + (lane & 15);
            v2f af;                              // contiguous -> ds_load_b64
            af.x = Wl[m * KD + ka];
            af.y = Wl[m * KD + ka + 1];
            acc[mt] = __builtin_amdgcn_wmma_f32_16x16x4_f32(
                false, af, false, bf, (short)0, acc[mt], false, false);
        }
    }

    #pragma unroll
    for (int mt = 0; mt < MT; ++mt) {
        #pragma unroll
        for (int r = 0; r < 8; ++r) {
            const int ch = mt * 16 + r + 8 * h;
            const float v = acc[mt][r] + bias[ch];
            if (TO_OUT) {
                const int b   = col / (S_ * KSL);
                const int rem = col - b * S_ * KSL;
                const int s   = rem / KSL;
                const int k   = rem % KSL;
                Y[(((size_t)b * 64 + ch) * S_ + s) * KSL + k] = v;
            } else {
                Y[(size_t)ch * P_ + col] = v;
            }
            atomicAdd(&red[slot + ch], v);        // ds_add_f32, conflict-free lanes
            atomicAdd(&red[slot + M + ch], v * v);
        }
    }
    __syncthreads();
    // fold the 16 slots per channel, then one global atomic per channel
    for (int i = tid; i < 2 * M; i += 256) {
        float sum = 0.0f;
        #pragma unroll
        for (int sl = 0; sl < 16; ++sl) sum += red[sl * 2 * M + i];
        atomicAdd(&stats[i], sum);                // global_atomic_add_f32
    }
}

// ---------------- BatchNorm (batch stats) + ReLU, in place on [C][P] ----------------
template <int C>
__global__ __launch_bounds__(256) void k_bnrelu(float* __restrict__ Y,
                                                const float* __restrict__ stats,
                                                const float* __restrict__ g,
                                                const float* __restrict__ be) {
    const int i  = blockIdx.x * 256 + threadIdx.x;   // over C*P_
    const int ch = i / P_;
    const float cnt   = (float)P_;
    const float mean  = stats[ch] / cnt;
    const float var   = stats[C + ch] / cnt - mean * mean;
    const float scale = g[ch] * rsqrtf(var + EPS_);
    const float shift = be[ch] - mean * scale;
    const float v = Y[i] * scale + shift;
    Y[i] = v > 0.0f ? v : 0.0f;
}

// ---------------- final BN + ReLU + empty-group mask, in place on d_out ----------------
__global__ __launch_bounds__(256) void k_final(float* __restrict__ out,
                                               const float* __restrict__ stats,
                                               const float* __restrict__ g,
                                               const float* __restrict__ be,
                                               const int* __restrict__ num) {
    const int i = blockIdx.x * 256 + threadIdx.x;    // over B*64*S*K
    const int s = (i / KSL) % S_;
    const int c = (i / (KSL * S_)) % 64;
    const int b = i / (KSL * S_ * 64);
    const float cnt   = (float)P_;
    const float mean  = stats[c] / cnt;
    const float var   = stats[64 + c] / cnt - mean * mean;
    const float scale = g[c] * rsqrtf(var + EPS_);
    const float shift = be[c] - mean * scale;
    float v = out[i] * scale + shift;
    v = v > 0.0f ? v : 0.0f;
    out[i] = (num[b * S_ + s] > 0) ? v : 0.0f;
}

// ---------------- host side ----------------
extern "C" void kernel_launch(void* const* d_in, const int* in_sizes, int n_in,
                              void* d_out, int out_size, void* d_ws, size_t ws_size,
                              hipStream_t stream) {
    const float* pc     = (const float*)d_in[0];
    const float* feat   = (const float*)d_in[1];
    const float* new_pc = (const float*)d_in[2];
    const float* w1  = (const float*)d_in[3];
    const float* b1  = (const float*)d_in[4];
    const float* g1  = (const float*)d_in[5];
    const float* be1 = (const float*)d_in[6];
    const float* w2  = (const float*)d_in[7];
    const float* b2  = (const float*)d_in[8];
    const float* g2  = (const float*)d_in[9];
    const float* be2 = (const float*)d_in[10];
    const float* w3  = (const float*)d_in[11];
    const float* b3  = (const float*)d_in[12];
    const float* g3  = (const float*)d_in[13];
    const float* be3 = (const float*)d_in[14];
    float* out = (float*)d_out;

    // workspace layout
    float* X0  = (float*)d_ws;                   // 20*P floats (row 19 zero-padded)
    float* Y1  = X0 + 20 * (size_t)P_;           // 32*P
    float* Y2  = Y1 + 32 * (size_t)P_;           // 32*P
    int*   idx = (int*)(Y2 + 32 * (size_t)P_);   // P ints
    int*   num = idx + P_;                       // B*S ints
    float* st1 = (float*)(num + B_ * S_);        // 128 floats
    float* st2 = st1 + 128;                      // 128 floats
    float* st3 = st2 + 128;                      // 128 floats

    k_init<<<(P_ + 255) / 256, 256, 0, stream>>>(X0, st1);
    k_ballquery<<<B_ * (S_ / 256), 256, 0, stream>>>(pc, new_pc, idx, num);
    k_gather<<<P_ / 256, 256, 0, stream>>>(pc, feat, new_pc, idx, X0);

    const int gemm_blocks = P_ / 128;            // 8 waves/block * 16 cols/wave
    k_gemm<2, 20, 19, false><<<gemm_blocks, 256, 0, stream>>>(X0, w1, b1, Y1, st1);
    k_bnrelu<32><<<(32 * P_) / 256, 256, 0, stream>>>(Y1, st1, g1, be1);
    k_gemm<2, 32, 32, false><<<gemm_blocks, 256, 0, stream>>>(Y1, w2, b2, Y2, st2);
    k_bnrelu<32><<<(32 * P_) / 256, 256, 0, stream>>>(Y2, st2, g2, be2);
    k_gemm<4, 32, 32, true><<<gemm_blocks, 256, 0, stream>>>(Y2, w3, b3, out, st3);
    k_final<<<(B_ * 64 * S_ * KSL) / 256, 256, 0, stream>>>(out, st3, g3, be3, num);
}